// HierarchicalModel_47991964566134
// MI455X (gfx1250) — compile-verified
//
#include <hip/hip_runtime.h>

typedef __attribute__((ext_vector_type(2))) float v2f;
typedef __attribute__((ext_vector_type(8))) float v8f;

// Problem sizes (fixed by the reference)
constexpr int Bc = 64, Tc = 1000, Dc = 80, Kc = 12, Pc = 100;
constexpr int PITCH = 81;                 // LDS pitch, odd -> conflict-free columns
constexpr int NTILE = (Tc + 15) / 16;     // 63 column tiles of 16
constexpr float DLOG2PI = 147.030165312747630f;  // D * log(2*pi)

// ---------------------------------------------------------------------------
// Kernel 1: present-subject mask
// ---------------------------------------------------------------------------
__global__ void hier_present_k(const int* __restrict__ sids, float* __restrict__ present) {
  int t = threadIdx.x;
  if (t < Pc) present[t] = 0.0f;
  __syncthreads();
  if (t < Bc) present[sids[t]] = 1.0f;  // racy but all writers store 1.0f
}

// ---------------------------------------------------------------------------
// Kernel 2: per-(b,k) weighted Gaussian NLL partial via FP32 WMMA GEMM
//   - invert L (80x80 lower-tri) into LDS (forward substitution per column)
//   - Y = Linv * R  with R[:,t] = x[b,t] - mu, via v_wmma_f32_16x16x4_f32
//   - mahal[t] = sum_d Y[d,t]^2 ; acc += gamma * (D*log2pi + logdet + mahal)
// ---------------------------------------------------------------------------
__global__ __launch_bounds__(256) void hier_nll_wmma_k(
    const float* __restrict__ x, const float* __restrict__ mu_subj,
    const float* __restrict__ L_subj, const float* __restrict__ gamma,
    const int* __restrict__ sids, float* __restrict__ nllpart) {
  __shared__ float Ll[Dc * PITCH];   // L (lower-tri)
  __shared__ float Al[Dc * PITCH];   // Linv (lower-tri, upper zeroed)
  __shared__ float mul[Dc];
  __shared__ float wacc[8];
  __shared__ float ldet_s;

  const int bk = blockIdx.x;
  const int b = bk / Kc;
  const int k = bk - b * Kc;
  const int tid = threadIdx.x;
  const int lane = tid & 31;
  const int wv = tid >> 5;
  const int sid = sids[b];

  // Stage L and mu into LDS (coalesced)
  const float* Lg = L_subj + ((size_t)sid * Kc + k) * (Dc * Dc);
  for (int i = tid; i < Dc * Dc; i += 256) {
    int r = i / Dc, c = i - r * Dc;
    Ll[r * PITCH + c] = Lg[i];
  }
  const float* mg = mu_subj + ((size_t)sid * Kc + k) * Dc;
  if (tid < Dc) mul[tid] = mg[tid];
  __syncthreads();

  // Triangular inversion: thread j computes column j of A = L^{-1}
  if (tid < Dc) {
    const int j = tid;
    for (int i = 0; i < j; ++i) Al[i * PITCH + j] = 0.0f;
    Al[j * PITCH + j] = 1.0f / Ll[j * PITCH + j];
    for (int i = j + 1; i < Dc; ++i) {
      float s = 0.0f;
      for (int m = j; m < i; ++m) s += Ll[i * PITCH + m] * Al[m * PITCH + j];
      Al[i * PITCH + j] = -s / Ll[i * PITCH + i];
    }
  } else if (tid == Dc) {
    float s = 0.0f;
    for (int i = 0; i < Dc; ++i) s += logf(Ll[i * PITCH + i]);
    ldet_s = 2.0f * s;
  }
  __syncthreads();

  const float ldet = ldet_s;
  const int tl = lane & 15;           // output column within tile (N)
  const int koff = (lane >> 4) << 1;  // K sub-offset: lanes 0-15 -> K=0,1 ; 16-31 -> K=2,3
  float acc = 0.0f;

  for (int tile = wv; tile < NTILE; tile += 8) {
    const int tbase = tile * 16;
    const int t = tbase + tl;
    const int tc = (t < Tc) ? t : (Tc - 1);  // clamp tail; gamma gate kills it
    const float* xrow = x + ((size_t)b * Tc + tc) * Dc;
    if (tile + 8 < NTILE) __builtin_prefetch(xrow + (size_t)8 * 16 * Dc, 0, 1);

    // 5 row-block accumulators: full 80 rows of Y for this 16-column tile
    v8f c0 = {}, c1 = {}, c2 = {}, c3 = {}, c4 = {};
    for (int kk = 0; kk < 20; ++kk) {
      const int kb = kk * 4 + koff;
      // B fragment (4x16): R[K=kb..kb+1, N=tl] = x - mu
      v2f bb;
      bb.x = xrow[kb] - mul[kb];
      bb.y = xrow[kb + 1] - mul[kb + 1];
      // A fragments (16x4) from LDS: A[m*16+tl, kb..kb+1]
      v2f a0, a1, a2, a3, a4;
      a0.x = Al[(0 * 16 + tl) * PITCH + kb]; a0.y = Al[(0 * 16 + tl) * PITCH + kb + 1];
      a1.x = Al[(1 * 16 + tl) * PITCH + kb]; a1.y = Al[(1 * 16 + tl) * PITCH + kb + 1];
      a2.x = Al[(2 * 16 + tl) * PITCH + kb]; a2.y = Al[(2 * 16 + tl) * PITCH + kb + 1];
      a3.x = Al[(3 * 16 + tl) * PITCH + kb]; a3.y = Al[(3 * 16 + tl) * PITCH + kb + 1];
      a4.x = Al[(4 * 16 + tl) * PITCH + kb]; a4.y = Al[(4 * 16 + tl) * PITCH + kb + 1];
      c0 = __builtin_amdgcn_wmma_f32_16x16x4_f32(false, a0, false, bb, (short)0, c0, false, false);
      c1 = __builtin_amdgcn_wmma_f32_16x16x4_f32(false, a1, false, bb, (short)0, c1, false, false);
      c2 = __builtin_amdgcn_wmma_f32_16x16x4_f32(false, a2, false, bb, (short)0, c2, false, false);
      c3 = __builtin_amdgcn_wmma_f32_16x16x4_f32(false, a3, false, bb, (short)0, c3, false, false);
      c4 = __builtin_amdgcn_wmma_f32_16x16x4_f32(false, a4, false, bb, (short)0, c4, false, false);
    }

    // Sum of squares of this lane's half-column (rows M=r(+8) across 5 blocks)
    float ms = 0.0f;
#pragma unroll
    for (int r = 0; r < 8; ++r) {
      ms += c0[r] * c0[r] + c1[r] * c1[r] + c2[r] * c2[r] + c3[r] * c3[r] + c4[r] * c4[r];
    }
    // Combine the two row-halves of column tl (lanes L and L+16)
    ms += __shfl_xor(ms, 16, 32);
    if (lane < 16 && t < Tc) {
      float g = gamma[((size_t)b * Tc + t) * Kc + k];
      acc += g * (DLOG2PI + ldet + ms);
    }
  }

  // Deterministic block reduction: wave shuffles, then fixed-order sum
  for (int off = 16; off > 0; off >>= 1) acc += __shfl_down(acc, off, 32);
  if (lane == 0) wacc[wv] = acc;
  __syncthreads();
  if (tid == 0) {
    float s = 0.0f;
    for (int w = 0; w < 8; ++w) s += wacc[w];
    nllpart[bk] = s;
  }
}

// ---------------------------------------------------------------------------
// Kernel 3: regularizer partials (grid-strided, deterministic per block)
// ---------------------------------------------------------------------------
__global__ __launch_bounds__(256) void hier_reg_k(
    const float* __restrict__ mu_pop, const float* __restrict__ L_pop,
    const float* __restrict__ mu_subj, const float* __restrict__ L_subj,
    const float* __restrict__ present, const float* __restrict__ lam_mu,
    const float* __restrict__ lam_L, float* __restrict__ regpart) {
  const size_t nmu = (size_t)Pc * Kc * Dc;            // 96,000
  const size_t nL = (size_t)Pc * Kc * Dc * Dc;        // 7,680,000
  const size_t stride = (size_t)gridDim.x * blockDim.x;
  const size_t i0 = (size_t)blockIdx.x * blockDim.x + threadIdx.x;

  float smu = 0.0f, sL = 0.0f;
  for (size_t i = i0; i < nmu; i += stride) {
    int p = (int)(i / (Kc * Dc));
    int rem = (int)(i - (size_t)p * (Kc * Dc));
    float d = (mu_subj[i] - mu_pop[rem]) * present[p];
    smu += d * d;
  }
  for (size_t i = i0; i < nL; i += stride) {
    int p = (int)(i / ((size_t)Kc * Dc * Dc));
    size_t rem = i - (size_t)p * ((size_t)Kc * Dc * Dc);
    float d = (L_subj[i] - L_pop[rem]) * present[p];
    sL += d * d;
  }
  __shared__ float sm[256];
  sm[threadIdx.x] = 0.5f * (lam_mu[0] * smu + lam_L[0] * sL);
  __syncthreads();
  for (int off = 128; off > 0; off >>= 1) {
    if ((int)threadIdx.x < off) sm[threadIdx.x] += sm[threadIdx.x + off];
    __syncthreads();
  }
  if (threadIdx.x == 0) regpart[blockIdx.x] = sm[0];
}

// ---------------------------------------------------------------------------
// Kernel 4: final deterministic combine -> scalar
// ---------------------------------------------------------------------------
__global__ __launch_bounds__(256) void hier_final_k(
    const float* __restrict__ nllpart, const float* __restrict__ regpart,
    const float* __restrict__ present, const int* __restrict__ n_subjects,
    float* __restrict__ out) {
  __shared__ float sm[256];
  const int t = threadIdx.x;

  sm[t] = nllpart[t] + nllpart[t + 256] + nllpart[t + 512];  // 768 partials
  __syncthreads();
  for (int off = 128; off > 0; off >>= 1) {
    if (t < off) sm[t] += sm[t + off];
    __syncthreads();
  }
  const float nll_total = sm[0];
  __syncthreads();

  sm[t] = regpart[t];  // 256 partials
  __syncthreads();
  for (int off = 128; off > 0; off >>= 1) {
    if (t < off) sm[t] += sm[t + off];
    __syncthreads();
  }
  if (t == 0) {
    float reg_total = sm[0];
    float cnt = 0.0f;
    for (int p = 0; p < Pc; ++p) cnt += present[p];
    const float scale = cnt / (float)n_subjects[0];
    out[0] = 0.5f * nll_total / (float)(Bc * Tc) + scale * reg_total;
  }
}

// ---------------------------------------------------------------------------
extern "C" void kernel_launch(void* const* d_in, const int* in_sizes, int n_in,
                              void* d_out, int out_size, void* d_ws, size_t ws_size,
                              hipStream_t stream) {
  (void)in_sizes; (void)n_in; (void)out_size; (void)ws_size;
  const float* x       = (const float*)d_in[0];
  const float* mu_pop  = (const float*)d_in[1];
  const float* L_pop   = (const float*)d_in[2];
  const float* mu_subj = (const float*)d_in[3];
  const float* L_subj  = (const float*)d_in[4];
  const float* gam     = (const float*)d_in[5];
  const int*   sids    = (const int*)d_in[6];
  const float* lam_mu  = (const float*)d_in[7];
  const float* lam_L   = (const float*)d_in[8];
  const int*   n_subj  = (const int*)d_in[9];
  float* out = (float*)d_out;

  float* ws = (float*)d_ws;
  float* nllpart = ws;         // 768 floats
  float* regpart = ws + 768;   // 256 floats
  float* present = ws + 1024;  // 100 floats

  hier_present_k<<<1, 128, 0, stream>>>(sids, present);
  hier_nll_wmma_k<<<Bc * Kc, 256, 0, stream>>>(x, mu_subj, L_subj, gam, sids, nllpart);
  hier_reg_k<<<256, 256, 0, stream>>>(mu_pop, L_pop, mu_subj, L_subj, present, lam_mu, lam_L, regpart);
  hier_final_k<<<1, 256, 0, stream>>>(nllpart, regpart, present, n_subj, out);
}